// GNNModel_79912161509825
// MI455X (gfx1250) — compile-verified
//
#include <hip/hip_runtime.h>

typedef __attribute__((ext_vector_type(2))) float v2f;
typedef __attribute__((ext_vector_type(8))) float v8f;

#define NN 100000
#define NE 1200000
#define CH 64
#define NTILES (NN / 16)   // 6250, exact

// ---------------- prep: nan_to_num(x) -> xc, zero agg/cnt ----------------
__global__ void k_clean_zero(const float* __restrict__ x, float* __restrict__ xc,
                             float* __restrict__ agg, float* __restrict__ cnt) {
    int i = blockIdx.x * blockDim.x + threadIdx.x;
    if (i < NN * CH) {
        float v = x[i];
        xc[i] = (v == v) ? v : 0.0f;   // nan -> 0
        agg[i] = 0.0f;
    }
    if (i < NN) cnt[i] = 0.0f;
}

__global__ void k_zero(float* __restrict__ agg, float* __restrict__ cnt) {
    int i = blockIdx.x * blockDim.x + threadIdx.x;
    if (i < NN * CH) agg[i] = 0.0f;
    if (i < NN) cnt[i] = 0.0f;
}

// ---------------- edge-parallel mean-aggregation scatter ----------------
// one thread per (edge, channel); atomics resolve in L2 (whole graph is L2-resident)
__global__ void k_scatter(const float* __restrict__ feat,
                          const int* __restrict__ src, const int* __restrict__ dst,
                          float* __restrict__ agg, float* __restrict__ cnt) {
    long long i = (long long)blockIdx.x * blockDim.x + threadIdx.x;
    if (i >= (long long)NE * CH) return;
    int e = (int)(i >> 6);
    int c = (int)(i & 63);
    int s = src[e];
    int d = dst[e];
    atomicAdd(&agg[d * CH + c], feat[s * CH + c]);
    if (c == 0) atomicAdd(&cnt[d], 1.0f);
}

// ---------------- SAGE GEMM: out = relu(mean @ Wl + bias + feat @ Wr) ----------------
// One wave32 per 16-node tile. A = [mean | feat] (16 x 128), W = [Wl ; Wr] (128 x 64).
// 32 K-steps of V_WMMA_F32_16X16X4_F32 across 4 N-tiles of 16 columns.
// Weights pre-swizzled into B-fragment order in LDS: Wfrag[s][nt][lane][v].
__global__ void __launch_bounds__(256)
k_sage_gemm(const float* __restrict__ agg, const float* __restrict__ cnt,
            const float* __restrict__ feat,
            const float* __restrict__ Wl, const float* __restrict__ Wr,
            const float* __restrict__ bias, float* __restrict__ out) {
    __shared__ float Wfrag[32 * 4 * 32 * 2];   // 8192 floats = 32 KB

    // cooperative swizzled weight load: element (s,nt,lane,v) = Wcat[4s + (lane>>4)*2 + v][nt*16 + (lane&15)]
    for (int idx = threadIdx.x; idx < 32 * 4 * 32 * 2; idx += 256) {
        int v  = idx & 1;
        int l  = (idx >> 1) & 31;
        int nt = (idx >> 6) & 3;
        int s  = idx >> 8;
        int K  = 4 * s + ((l >> 4) << 1) + v;
        int n  = nt * 16 + (l & 15);
        Wfrag[idx] = (K < 64) ? Wl[K * 64 + n] : Wr[(K - 64) * 64 + n];
    }
    __syncthreads();

    int lane = threadIdx.x & 31;
    int hi   = lane >> 4;        // 0 or 1
    int r    = lane & 15;
    int tile = blockIdx.x * 8 + (threadIdx.x >> 5);
    bool active = (tile < NTILES);   // wave-uniform: EXEC stays all-1s inside

    if (active) {
        int row = tile * 16 + r;
        float rd = 1.0f / fmaxf(cnt[row], 1.0f);
        const float* aggrow  = agg  + row * CH;
        const float* featrow = feat + row * CH;

        v8f acc[4];
        #pragma unroll
        for (int nt = 0; nt < 4; ++nt) {
            float bv = bias[nt * 16 + r];
            #pragma unroll
            for (int j = 0; j < 8; ++j) acc[nt][j] = bv;
        }

        #pragma unroll 4
        for (int s = 0; s < 32; ++s) {
            // A fragment: lane holds A[M = r][K = 4s + hi*2 + {0,1}]
            int kloc = 4 * (s & 15) + hi * 2;
            const float* base = (s < 16) ? aggrow : featrow;
            v2f a = *(const v2f*)(base + kloc);
            if (s < 16) { a.x *= rd; a.y *= rd; }

            #pragma unroll
            for (int nt = 0; nt < 4; ++nt) {
                v2f b = *(const v2f*)&Wfrag[((s * 4 + nt) * 32 + lane) * 2];
                // (neg_a, A, neg_b, B, c_mod, C, reuse_a, reuse_b)
                acc[nt] = __builtin_amdgcn_wmma_f32_16x16x4_f32(
                    false, a, false, b, (short)0, acc[nt], false, false);
            }
        }

        // D layout: lane holds D[M = v + hi*8][N = nt*16 + r]
        #pragma unroll
        for (int nt = 0; nt < 4; ++nt) {
            #pragma unroll
            for (int v = 0; v < 8; ++v) {
                int orow = tile * 16 + hi * 8 + v;
                out[orow * CH + nt * 16 + r] = fmaxf(acc[nt][v], 0.0f);
            }
        }
    }
}

// ---------------- final FC: pred = h2 @ Wfc + bfc, wave per node ----------------
__global__ void k_fc(const float* __restrict__ h2, const float* __restrict__ Wfc,
                     const float* __restrict__ bfc, float* __restrict__ pred) {
    int node = blockIdx.x * 8 + (threadIdx.x >> 5);
    if (node >= NN) return;
    int lane = threadIdx.x & 31;
    const float* hrow = h2 + node * CH;
    float s = hrow[lane] * Wfc[lane] + hrow[lane + 32] * Wfc[lane + 32];
    #pragma unroll
    for (int off = 16; off > 0; off >>= 1) s += __shfl_down(s, off, 32);
    if (lane == 0) pred[node] = s + bfc[0];
}

extern "C" void kernel_launch(void* const* d_in, const int* in_sizes, int n_in,
                              void* d_out, int out_size, void* d_ws, size_t ws_size,
                              hipStream_t stream) {
    const float* x   = (const float*)d_in[0];
    const int*   ei  = (const int*)d_in[1];     // edge_index [2, E]
    const float* W1l = (const float*)d_in[2];
    const float* b1  = (const float*)d_in[3];
    const float* W1r = (const float*)d_in[4];
    const float* W2l = (const float*)d_in[5];
    const float* b2  = (const float*)d_in[6];
    const float* W2r = (const float*)d_in[7];
    const float* Wfc = (const float*)d_in[8];
    const float* bfc = (const float*)d_in[9];
    float* pred = (float*)d_out;

    const int* src = ei;
    const int* dst = ei + NE;

    // workspace carve-up (77.2 MB total)
    float* xc  = (float*)d_ws;          // N*64
    float* agg = xc  + NN * CH;         // N*64
    float* cnt = agg + NN * CH;         // N
    float* h1  = cnt + NN;              // N*64
    float* h2  = xc;                    // reuse: xc dead after layer-1 gemm

    const int zgrid = (NN * CH + 255) / 256;                       // 25000
    const int sgrid = (int)(((long long)NE * CH + 255) / 256);     // 300000
    const int ggrid = (NTILES + 7) / 8;                            // 782
    const int fgrid = (NN + 7) / 8;                                // 12500

    // layer 1
    k_clean_zero<<<zgrid, 256, 0, stream>>>(x, xc, agg, cnt);
    k_scatter<<<sgrid, 256, 0, stream>>>(xc, src, dst, agg, cnt);
    k_sage_gemm<<<ggrid, 256, 0, stream>>>(agg, cnt, xc, W1l, W1r, b1, h1);

    // layer 2
    k_zero<<<zgrid, 256, 0, stream>>>(agg, cnt);
    k_scatter<<<sgrid, 256, 0, stream>>>(h1, src, dst, agg, cnt);
    k_sage_gemm<<<ggrid, 256, 0, stream>>>(agg, cnt, h1, W2l, W2r, b2, h2);

    // head
    k_fc<<<fgrid, 256, 0, stream>>>(h2, Wfc, bfc, pred);
}